// RoIPool_53575422050620
// MI455X (gfx1250) — compile-verified
//
#include <hip/hip_runtime.h>
#include <math.h>

typedef float v4f __attribute__((ext_vector_type(4)));
typedef int   v4i __attribute__((ext_vector_type(4)));

// ---------------------------------------------------------------------------
// Phase 1: one thread per output position. Does ALL the per-position math
// (fp32 divisions, floor, clips, weights) exactly once and emits a 32-byte
// descriptor: { o00, o01, o10, o11, w00, w01, w10, w11 }.
// 98k threads total -> negligible cost; descriptors = 3.1 MB (L2-resident).
// ---------------------------------------------------------------------------
template <int PS>
__global__ __launch_bounds__(256) void roi_prep_kernel(
    const float* __restrict__ rois,  // [N,4] = x,y,w,h (float, used as int)
    int* __restrict__ desc,          // [total][8]
    int total, int H, int W, int ps_rt)
{
    constexpr int C = 256;
    int idx = blockIdx.x * 256 + threadIdx.x;
    if (idx >= total) return;

    const int ps = (PS > 0) ? PS : ps_rt;
    const int pp = ps * ps;
    int n   = idx / pp;              // mul-shift when PS known at compile time
    int rem = idx - n * pp;
    int py  = rem / ps;
    int px  = rem - py * ps;

    v4f r = reinterpret_cast<const v4f*>(rois)[n];
    int x = (int)r[0], y = (int)r[1], w = (int)r[2], h = (int)r[3];

    // Bilinear sample coordinates (matches reference math)
    float sy   = (float)h / (float)ps;
    float sx   = (float)w / (float)ps;
    float srcy = (float)py * sy;
    float srcx = (float)px * sx;
    int   y0   = (int)floorf(srcy);
    int   x0   = (int)floorf(srcx);
    float wy   = srcy - (float)y0;
    float wx   = srcx - (float)x0;

    int gy0 = min(max(y + min(max(y0,     0), h - 1), 0), H - 1);
    int gy1 = min(max(y + min(max(y0 + 1, 0), h - 1), 0), H - 1);
    int gx0 = min(max(x + min(max(x0,     0), w - 1), 0), W - 1);
    int gx1 = min(max(x + min(max(x0 + 1, 0), w - 1), 0), W - 1);

    v4i offs;
    offs[0] = (gy0 * W + gx0) * C;
    offs[1] = (gy0 * W + gx1) * C;
    offs[2] = (gy1 * W + gx0) * C;
    offs[3] = (gy1 * W + gx1) * C;

    v4f wts;
    wts[0] = (1.0f - wy) * (1.0f - wx);
    wts[1] = (1.0f - wy) * wx;
    wts[2] = wy * (1.0f - wx);
    wts[3] = wy * wx;

    int* d = desc + (size_t)idx * 8;
    *reinterpret_cast<v4i*>(d)     = offs;
    *reinterpret_cast<v4f*>(d + 4) = wts;
}

// ---------------------------------------------------------------------------
// Phase 2: one wave32 per output position, 8 channels per lane (256 total).
// Position index is forced wave-uniform via readfirstlane so the descriptor
// is fetched with scalar loads and tap bases live in SGPRs; lanes contribute
// only the channel offset. 8x b128 coalesced tap loads (L2 hits: the 41 MB
// feature map fits the 192 MB L2) + 2x non-temporal b128 stores so the
// 100 MB streaming output doesn't evict the feature map.
// ---------------------------------------------------------------------------
__global__ __launch_bounds__(256) void roi_gather_kernel(
    const float* __restrict__ img,   // [H,W,C]
    const int*   __restrict__ desc,  // [total][8]
    float* __restrict__ out,         // [N,ps,ps,C]
    int total)
{
    constexpr int C = 256;
    int tid = blockIdx.x * 256 + threadIdx.x;
    int idx = tid >> 5;                      // one wave per position
    if (idx >= total) return;                // wave-uniform exit
    idx = __builtin_amdgcn_readfirstlane(idx);
    int lane = threadIdx.x & 31;

    const int*   d  = desc + (size_t)idx * 8;            // uniform -> s_load
    const float* df = reinterpret_cast<const float*>(d);
    int   o00 = d[0],  o01 = d[1],  o10 = d[2],  o11 = d[3];
    float w00 = df[4], w01 = df[5], w10 = df[6], w11 = df[7];

    int c = lane << 3;  // 8 channels per lane
    const v4f* p00 = reinterpret_cast<const v4f*>(img + o00 + c);
    const v4f* p01 = reinterpret_cast<const v4f*>(img + o01 + c);
    const v4f* p10 = reinterpret_cast<const v4f*>(img + o10 + c);
    const v4f* p11 = reinterpret_cast<const v4f*>(img + o11 + c);

    // 8 independent b128 loads in flight
    v4f a0 = p00[0], a1 = p00[1];
    v4f b0 = p01[0], b1 = p01[1];
    v4f c0 = p10[0], c1 = p10[1];
    v4f e0 = p11[0], e1 = p11[1];

    v4f olo = a0 * w00 + b0 * w01 + c0 * w10 + e0 * w11;
    v4f ohi = a1 * w00 + b1 * w01 + c1 * w10 + e1 * w11;

    float* op = out + (size_t)idx * C + c;
    __builtin_nontemporal_store(olo, reinterpret_cast<v4f*>(op));
    __builtin_nontemporal_store(ohi, reinterpret_cast<v4f*>(op + 4));
}

// ---------------------------------------------------------------------------
// Fallback: fused single kernel (used only if d_ws is too small).
// ---------------------------------------------------------------------------
template <int PS>
__global__ __launch_bounds__(256) void roi_bilinear_fused(
    const float* __restrict__ img, const float* __restrict__ rois,
    float* __restrict__ out, int total, int H, int W, int ps_rt)
{
    constexpr int C = 256;
    int tid = blockIdx.x * 256 + threadIdx.x;
    int idx = tid >> 6;
    if (idx >= total) return;
    int q = tid & 63;

    const int ps = (PS > 0) ? PS : ps_rt;
    const int pp = ps * ps;
    int n   = idx / pp;
    int rem = idx - n * pp;
    int py  = rem / ps;
    int px  = rem - py * ps;

    v4f r = reinterpret_cast<const v4f*>(rois)[n];
    int x = (int)r[0], y = (int)r[1], w = (int)r[2], h = (int)r[3];

    float sy   = (float)h / (float)ps;
    float sx   = (float)w / (float)ps;
    float srcy = (float)py * sy;
    float srcx = (float)px * sx;
    int   y0   = (int)floorf(srcy);
    int   x0   = (int)floorf(srcx);
    float wy   = srcy - (float)y0;
    float wx   = srcx - (float)x0;

    int gy0 = min(max(y + min(max(y0,     0), h - 1), 0), H - 1);
    int gy1 = min(max(y + min(max(y0 + 1, 0), h - 1), 0), H - 1);
    int gx0 = min(max(x + min(max(x0,     0), w - 1), 0), W - 1);
    int gx1 = min(max(x + min(max(x0 + 1, 0), w - 1), 0), W - 1);

    int c = q << 2;
    v4f v00 = *reinterpret_cast<const v4f*>(img + ((size_t)(gy0 * W + gx0) * C + c));
    v4f v01 = *reinterpret_cast<const v4f*>(img + ((size_t)(gy0 * W + gx1) * C + c));
    v4f v10 = *reinterpret_cast<const v4f*>(img + ((size_t)(gy1 * W + gx0) * C + c));
    v4f v11 = *reinterpret_cast<const v4f*>(img + ((size_t)(gy1 * W + gx1) * C + c));

    float w00 = (1.0f - wy) * (1.0f - wx);
    float w01 = (1.0f - wy) * wx;
    float w10 = wy * (1.0f - wx);
    float w11 = wy * wx;

    v4f o = v00 * w00 + v01 * w01 + v10 * w10 + v11 * w11;
    __builtin_nontemporal_store(o, reinterpret_cast<v4f*>(out + (size_t)idx * C + c));
}

extern "C" void kernel_launch(void* const* d_in, const int* in_sizes, int n_in,
                              void* d_out, int out_size, void* d_ws, size_t ws_size,
                              hipStream_t stream) {
    const float* img  = (const float*)d_in[0];  // (1,H,W,C) fp32
    const float* rois = (const float*)d_in[1];  // (1,N,4)   fp32
    float* out = (float*)d_out;                 // (1,N,ps,ps,C) fp32

    constexpr int C = 256;
    int N  = in_sizes[1] / 4;
    int hw = in_sizes[0] / C;                   // H*W (square per reference)
    int H  = (int)(sqrt((double)hw) + 0.5);
    int W  = H;
    int pp = (N > 0) ? (out_size / (N * C)) : 49;
    int ps = (int)(sqrt((double)pp) + 0.5);
    if (ps < 1) ps = 1;

    int total = N * ps * ps;                    // output positions
    size_t desc_bytes = (size_t)total * 8 * sizeof(int);

    if (ws_size >= desc_bytes) {
        int* desc = (int*)d_ws;
        // Phase 1: per-position math, once.
        dim3 pb(256), pg((total + 255) / 256);
        if (ps == 7)
            roi_prep_kernel<7><<<pg, pb, 0, stream>>>(rois, desc, total, H, W, ps);
        else
            roi_prep_kernel<0><<<pg, pb, 0, stream>>>(rois, desc, total, H, W, ps);
        // Phase 2: one wave per position, pure data movement.
        long long threads = (long long)total * 32;
        dim3 gb(256), gg((unsigned)((threads + 255) / 256));
        roi_gather_kernel<<<gg, gb, 0, stream>>>(img, desc, out, total);
    } else {
        long long threads = (long long)total * 64;
        dim3 block(256), grid((unsigned)((threads + 255) / 256));
        if (ps == 7)
            roi_bilinear_fused<7><<<grid, block, 0, stream>>>(img, rois, out, total, H, W, ps);
        else
            roi_bilinear_fused<0><<<grid, block, 0, stream>>>(img, rois, out, total, H, W, ps);
    }
}